// GAT_56908316672287
// MI455X (gfx1250) — compile-verified
//
#include <hip/hip_runtime.h>

// ---------------------------------------------------------------------------
// GAT (2-layer) for MI455X / gfx1250, wave32.
//  - fp32 WMMA (v_wmma_f32_16x16x4_f32) for both dense GEMMs
//  - weight panel staged to LDS via global_load_async_to_lds_b128 (ASYNCcnt)
//  - sparse per-row softmax (never materializes the dense 8192x8192 adjacency)
// ---------------------------------------------------------------------------

#define NNODE   8192
#define NFEAT   512
#define NHID    64
#define NHEADS  4
#define NCLASS  16
#define NEDGE   262144
#define ALPHA   0.2f
#define LDS_PAD 4              // floats of row padding -> bank-conflict-free B reads

typedef __attribute__((ext_vector_type(2))) float v2f;
typedef __attribute__((ext_vector_type(8))) float v8f;

// Monotone float<->u32 encoding so atomicMax(u32) implements float max.
__device__ __forceinline__ unsigned fenc(float x) {
    unsigned u = __float_as_uint(x);
    return (u & 0x80000000u) ? ~u : (u | 0x80000000u);
}
__device__ __forceinline__ float fdec(unsigned e) {
    return __uint_as_float((e & 0x80000000u) ? (e & 0x7FFFFFFFu) : ~e);
}

// ---------------------------------------------------------------------------
// Dense GEMM via fp32 WMMA with LDS-staged B panel.
//   H[head][m][n] = sum_k A[m][k] * W[head][n][k] + b[head][n]
// Block = 128 threads (4 waves). Block covers a 64-row strip of A and one
// 16-col panel of W.  The W panel (16 x K) is copied to LDS once per block
// with global_load_async_to_lds_b128 (tracked by ASYNCcnt), then all 4 waves
// read their B fragments from LDS (ds_load_b64) while streaming A from
// global (with global_prefetch_b8 lookahead).
// ---------------------------------------------------------------------------
__global__ void __launch_bounds__(128)
k_gemm_wmma(const float* __restrict__ A,     // [M x K] row-major
            const float* __restrict__ W,     // [heads][ncols][K] row-major
            const float* __restrict__ bias,  // [heads][ncols]
            float* __restrict__ H,           // [heads][M][ncols]
            int K, int ncols,
            long long wStrideHead, long long bStrideHead, long long hStrideHead)
{
    extern __shared__ float sB[];            // [16][K + LDS_PAD]
    const int tid  = threadIdx.x;
    const int lane = tid & 31;
    const int wave = tid >> 5;

    const int n0   = blockIdx.y * 16;
    const int head = blockIdx.z;

    const float* Wh = W    + (long long)head * wStrideHead + (long long)n0 * K;
    const float* bh = bias + (long long)head * bStrideHead;
    float*       Hh = H    + (long long)head * hStrideHead;

    // ---- async-stage the 16 x K weight panel into LDS (padded rows) ----
    {
        const int kv   = K >> 2;             // float4 chunks per row
        const int nvec = 16 * kv;
        for (int i = tid; i < nvec; i += 128) {
            const int r = i / kv;
            const int c = (i - r * kv) << 2;
            unsigned long long gaddr =
                (unsigned long long)(const void*)(Wh + (long long)r * K + c);
            unsigned ldsoff = (unsigned)((r * (K + LDS_PAD) + c) * 4);
            asm volatile("global_load_async_to_lds_b128 %0, %1, off"
                         :: "v"(ldsoff), "v"(gaddr) : "memory");
        }
        asm volatile("s_wait_asynccnt 0x0" ::: "memory");
        __syncthreads();
    }

    // ---- WMMA main loop: A from global, B from LDS ----
    const int m0   = blockIdx.x * 64 + wave * 16;
    const int row  = m0 + (lane & 15);
    const int cloc = lane & 15;              // local column inside the panel
    const int koff = (lane >> 4) * 2;        // lanes 16-31 carry K+2..K+3

    const float* aptr = A  + (long long)row * K + koff;
    const float* bld  = sB + cloc * (K + LDS_PAD) + koff;

    v8f c = {};
    for (int k = 0; k < K; k += 4) {
        __builtin_prefetch((const void*)(aptr + k + 64), 0, 1);  // global_prefetch_b8
        v2f a = *(const v2f*)(aptr + k);
        v2f b = *(const v2f*)(bld + k);                          // ds_load_b64
        c = __builtin_amdgcn_wmma_f32_16x16x4_f32(
                /*neg_a=*/false, a, /*neg_b=*/false, b,
                /*c_mod=*/(short)0, c, /*reuse_a=*/false, /*reuse_b=*/false);
    }

    const float bv    = bh[n0 + cloc];
    const int   rbase = m0 + (lane >> 4) * 8;  // C layout: lanes 16-31 hold M+8
#pragma unroll
    for (int i = 0; i < 8; ++i) {
        Hh[(long long)(rbase + i) * ncols + (n0 + cloc)] = c[i] + bv;
    }
}

// ---------------------------------------------------------------------------
// Per-node attention half-scores: ssrc[n] = H[n]·a[:dim], stgt[n] = H[n]·a[dim:]
// ---------------------------------------------------------------------------
__global__ void k_scores(const float* __restrict__ H, const float* __restrict__ a,
                         float* __restrict__ ssrc, float* __restrict__ stgt,
                         int nnode, int dim,
                         long long hStrideHead, long long aStrideHead, long long total)
{
    long long t = (long long)blockIdx.x * blockDim.x + threadIdx.x;
    if (t >= total) return;
    int head = (int)(t / nnode);
    int n    = (int)(t % nnode);
    const float* h  = H + (long long)head * hStrideHead + (long long)n * dim;
    const float* av = a + (long long)head * aStrideHead;
    float s0 = 0.f, s1 = 0.f;
    for (int d = 0; d < dim; ++d) {
        float hv = h[d];
        s0 = fmaf(hv, av[d], s0);
        s1 = fmaf(hv, av[dim + d], s1);
    }
    ssrc[t] = s0;
    stgt[t] = s1;
}

// ---------------------------------------------------------------------------
// Pass 1 over edges: per-source-row running max of leaky_relu(ssrc+stgt).
// ---------------------------------------------------------------------------
__global__ void k_edge_max(const int* __restrict__ src, const int* __restrict__ tgt,
                           const float* __restrict__ ssrc, const float* __restrict__ stgt,
                           unsigned* __restrict__ rowmax, int nnode, long long total)
{
    long long t = (long long)blockIdx.x * blockDim.x + threadIdx.x;
    if (t >= total) return;
    int head = (int)(t / NEDGE);
    long long e = t % NEDGE;
    int s = src[e], g = tgt[e];
    float v = ssrc[(long long)head * nnode + s] + stgt[(long long)head * nnode + g];
    v = (v > 0.f) ? v : ALPHA * v;
    atomicMax(&rowmax[(long long)head * nnode + s], fenc(v));
}

// ---------------------------------------------------------------------------
// Pass 2 over edges: num[src][:] += w_e * H[tgt][:],  den[src] += w_e,
// with w_e = exp(e - rowmax[src]). `dim` consecutive threads own one
// (edge,head) so H reads and atomics are coalesced.
// ---------------------------------------------------------------------------
__global__ void k_edge_agg(const int* __restrict__ src, const int* __restrict__ tgt,
                           const float* __restrict__ ssrc, const float* __restrict__ stgt,
                           const unsigned* __restrict__ rowmax,
                           const float* __restrict__ H,
                           float* __restrict__ num, float* __restrict__ den,
                           int nnode, int dim, long long totalEH)
{
    long long gid = (long long)blockIdx.x * blockDim.x + threadIdx.x;
    int d = (int)(gid % dim);
    long long eh = gid / dim;
    if (eh >= totalEH) return;
    int head = (int)(eh / NEDGE);
    long long e = eh % NEDGE;
    int s = src[e], g = tgt[e];
    long long srow = (long long)head * nnode + s;
    long long grow = (long long)head * nnode + g;
    float v = ssrc[srow] + stgt[grow];
    v = (v > 0.f) ? v : ALPHA * v;
    float w = __expf(v - fdec(rowmax[srow]));
    if (d == 0) atomicAdd(&den[srow], w);
    atomicAdd(&num[srow * dim + d], w * H[grow * dim + d]);
}

// ---------------------------------------------------------------------------
// Layer-1 finalize: out = num/den per head, ELU, concat heads -> F[n][256].
// ---------------------------------------------------------------------------
__global__ void k_final1(const float* __restrict__ num, const float* __restrict__ den,
                         float* __restrict__ F, int nnode, int dim, int nheads)
{
    long long t = (long long)blockIdx.x * blockDim.x + threadIdx.x;
    long long total = (long long)nnode * dim * nheads;
    if (t >= total) return;
    int c = (int)(t % (dim * nheads));
    int n = (int)(t / (dim * nheads));
    int head = c / dim;
    int d    = c % dim;
    long long row = (long long)head * nnode + n;
    float dd = den[row];
    float o  = (dd > 0.f) ? num[row * dim + d] / dd : 0.f;
    F[(long long)n * (dim * nheads) + head * dim + d] = (o > 0.f) ? o : (__expf(o) - 1.f);
}

// ---------------------------------------------------------------------------
// Layer-2 finalize: divide, then log_softmax over the 16 classes.
// ---------------------------------------------------------------------------
__global__ void k_final2(const float* __restrict__ num, const float* __restrict__ den,
                         float* __restrict__ out, int nnode)
{
    int n = blockIdx.x * blockDim.x + threadIdx.x;
    if (n >= nnode) return;
    float dd = den[n];
    float o[NCLASS];
    float m = -3.0e38f;
#pragma unroll
    for (int c = 0; c < NCLASS; ++c) {
        o[c] = (dd > 0.f) ? num[(long long)n * NCLASS + c] / dd : 0.f;
        m = fmaxf(m, o[c]);
    }
    float s = 0.f;
#pragma unroll
    for (int c = 0; c < NCLASS; ++c) s += __expf(o[c] - m);
    float ls = __logf(s);
#pragma unroll
    for (int c = 0; c < NCLASS; ++c) out[(long long)n * NCLASS + c] = o[c] - m - ls;
}

__global__ void k_fill_u(unsigned* __restrict__ p, unsigned v, long long nelem)
{
    long long i = (long long)blockIdx.x * blockDim.x + threadIdx.x;
    if (i < nelem) p[i] = v;
}

// ---------------------------------------------------------------------------
extern "C" void kernel_launch(void* const* d_in, const int* in_sizes, int n_in,
                              void* d_out, int out_size, void* d_ws, size_t ws_size,
                              hipStream_t stream)
{
    (void)in_sizes; (void)n_in; (void)out_size; (void)ws_size;

    const float* x     = (const float*)d_in[0];   // [8192, 512]
    const int*   edges = (const int*)  d_in[1];   // [2, E] (jax default int32)
    const float* W1    = (const float*)d_in[2];   // [4, 64, 512]
    const float* b1    = (const float*)d_in[3];   // [4, 64]
    const float* a1    = (const float*)d_in[4];   // [4, 128]
    const float* W2    = (const float*)d_in[5];   // [16, 256]
    const float* b2    = (const float*)d_in[6];   // [16]
    const float* a2    = (const float*)d_in[7];   // [32]
    float*       out   = (float*)d_out;           // [8192, 16]

    const int* src = edges;
    const int* tgt = edges + NEDGE;

    // ----- workspace carving (floats) -----
    float* ws    = (float*)d_ws;
    float* H1    = ws;                                        // 4*8192*64
    float* num1  = H1   + (long long)NHEADS * NNODE * NHID;
    float* F     = num1 + (long long)NHEADS * NNODE * NHID;   // 8192*256
    float* H2    = F    + (long long)NNODE * NHID * NHEADS;   // 8192*16
    float* num2  = H2   + (long long)NNODE * NCLASS;
    float* ssrc1 = num2 + (long long)NNODE * NCLASS;          // 4*8192
    float* stgt1 = ssrc1 + (long long)NHEADS * NNODE;
    float* den1  = stgt1 + (long long)NHEADS * NNODE;
    unsigned* rmax1 = (unsigned*)(den1 + (long long)NHEADS * NNODE);
    float* ssrc2 = (float*)(rmax1 + (long long)NHEADS * NNODE);
    float* stgt2 = ssrc2 + NNODE;
    float* den2  = stgt2 + NNODE;
    unsigned* rmax2 = (unsigned*)(den2 + NNODE);

    const unsigned ENC_NEG_INF = 0x007FFFFFu;  // fenc(-inf)

    // ----- init accumulators -----
    hipMemsetAsync(num1, 0, (size_t)NHEADS * NNODE * NHID * sizeof(float), stream);
    hipMemsetAsync(den1, 0, (size_t)NHEADS * NNODE * sizeof(float), stream);
    hipMemsetAsync(num2, 0, (size_t)NNODE * NCLASS * sizeof(float), stream);
    hipMemsetAsync(den2, 0, (size_t)NNODE * sizeof(float), stream);
    {
        long long n1 = (long long)NHEADS * NNODE;
        k_fill_u<<<(unsigned)((n1 + 255) / 256), 256, 0, stream>>>(rmax1, ENC_NEG_INF, n1);
        k_fill_u<<<(unsigned)((NNODE + 255) / 256), 256, 0, stream>>>(rmax2, ENC_NEG_INF, (long long)NNODE);
    }

    // ===== Layer 1 =====
    // H1 = x @ W1^T + b1   (per head); LDS panel = 16*(512+4)*4 bytes
    {
        dim3 grid(NNODE / 64, NHID / 16, NHEADS);
        size_t shmem = (size_t)16 * (NFEAT + LDS_PAD) * sizeof(float);
        k_gemm_wmma<<<grid, 128, shmem, stream>>>(
            x, W1, b1, H1, NFEAT, NHID,
            (long long)NHID * NFEAT, (long long)NHID, (long long)NNODE * NHID);
    }
    // per-node half scores
    {
        long long total = (long long)NHEADS * NNODE;
        k_scores<<<(unsigned)((total + 255) / 256), 256, 0, stream>>>(
            H1, a1, ssrc1, stgt1, NNODE, NHID,
            (long long)NNODE * NHID, (long long)2 * NHID, total);
    }
    // edge softmax max
    {
        long long total = (long long)NHEADS * NEDGE;
        k_edge_max<<<(unsigned)((total + 255) / 256), 256, 0, stream>>>(
            src, tgt, ssrc1, stgt1, rmax1, NNODE, total);
    }
    // edge weighted aggregation
    {
        long long totalEH = (long long)NHEADS * NEDGE;
        long long threads = totalEH * NHID;
        k_edge_agg<<<(unsigned)((threads + 255) / 256), 256, 0, stream>>>(
            src, tgt, ssrc1, stgt1, rmax1, H1, num1, den1, NNODE, NHID, totalEH);
    }
    // normalize + ELU + concat
    {
        long long total = (long long)NNODE * NHID * NHEADS;
        k_final1<<<(unsigned)((total + 255) / 256), 256, 0, stream>>>(
            num1, den1, F, NNODE, NHID, NHEADS);
    }

    // ===== Layer 2 =====
    // H2 = F @ W2^T + b2; LDS panel = 16*(256+4)*4 bytes
    {
        dim3 grid(NNODE / 64, NCLASS / 16, 1);
        size_t shmem = (size_t)16 * (NHID * NHEADS + LDS_PAD) * sizeof(float);
        k_gemm_wmma<<<grid, 128, shmem, stream>>>(
            F, W2, b2, H2, NHID * NHEADS, NCLASS, 0LL, 0LL, 0LL);
    }
    {
        long long total = NNODE;
        k_scores<<<(unsigned)((total + 255) / 256), 256, 0, stream>>>(
            H2, a2, ssrc2, stgt2, NNODE, NCLASS, 0LL, 0LL, total);
    }
    {
        long long total = NEDGE;
        k_edge_max<<<(unsigned)((total + 255) / 256), 256, 0, stream>>>(
            src, tgt, ssrc2, stgt2, rmax2, NNODE, total);
    }
    {
        long long totalEH = NEDGE;
        long long threads = totalEH * NCLASS;
        k_edge_agg<<<(unsigned)((threads + 255) / 256), 256, 0, stream>>>(
            src, tgt, ssrc2, stgt2, rmax2, H2, num2, den2, NNODE, NCLASS, totalEH);
    }
    // divide + log_softmax -> out
    k_final2<<<(NNODE + 255) / 256, 256, 0, stream>>>(num2, den2, out, NNODE);
}